// RobustPrompt_I_Feat_43490838839381
// MI455X (gfx1250) — compile-verified
//
#include <hip/hip_runtime.h>
#include <math.h>

#define C 128

typedef __attribute__((ext_vector_type(2))) float v2f;
typedef __attribute__((ext_vector_type(8))) float v8f;

// ---------------------------------------------------------------------------
// Kernel 1: per-node inverse L2 norm. One wave (32 lanes) per node,
// each lane loads a float4 (32*4 = 128 = C), butterfly-reduce, lane 0 writes.
// ---------------------------------------------------------------------------
__global__ void norm_kernel(const float* __restrict__ x, float* __restrict__ invn, int n) {
    int wave = (blockIdx.x * blockDim.x + threadIdx.x) >> 5;
    int lane = threadIdx.x & 31;
    if (wave >= n) return;
    float4 v = ((const float4*)(x + (size_t)wave * C))[lane];
    float s = v.x * v.x + v.y * v.y + v.z * v.z + v.w * v.w;
    #pragma unroll
    for (int off = 16; off; off >>= 1) s += __shfl_xor(s, off, 32);
    if (lane == 0) invn[wave] = rsqrtf(s);
}

// ---------------------------------------------------------------------------
// Kernel 2: edge phase. One wave per edge: gather both 512B rows (L2-resident,
// x fits in the 192MB L2), 128-dim dot via float4 + wave reduction, then
// scatter with global float atomics into c[col] and deg[col].
// x_norm[r].x_norm[c] == (x[r].x[c]) * invn[r] * invn[c]  (avoids storing x_norm)
// ---------------------------------------------------------------------------
__global__ void edge_kernel(const float* __restrict__ x, const int* __restrict__ ei,
                            const float* __restrict__ invn, float* __restrict__ csum,
                            float* __restrict__ deg, int nedges) {
    int e = (blockIdx.x * blockDim.x + threadIdx.x) >> 5;
    int lane = threadIdx.x & 31;
    if (e >= nedges) return;
    int row = ei[e];
    int col = ei[nedges + e];
    float4 a = ((const float4*)(x + (size_t)row * C))[lane];
    float4 b = ((const float4*)(x + (size_t)col * C))[lane];
    float d = a.x * b.x + a.y * b.y + a.z * b.z + a.w * b.w;
    #pragma unroll
    for (int off = 16; off; off >>= 1) d += __shfl_xor(d, off, 32);
    if (lane == 0) {
        atomicAdd(&csum[col], d * invn[row] * invn[col]);
        atomicAdd(&deg[col], 1.0f);
    }
}

// ---------------------------------------------------------------------------
// Kernel 3: compute the 4 possible attention outputs ("delta" vectors).
// The attention input rec depends on the node only via (sim_mask, deg_mask),
// i.e. 4 cases x 4 tokens = 16 rows -> ONE 16x384 GEMM with K=128, executed
// by a single wave with V_WMMA_F32_16X16X4_F32 (24 N-tiles x 32 K-steps).
//
// WMMA f32 16x16x4 layouts (ISA 7.12.2):
//   A (16x4): lane&15 = M row;  VGPR v holds K = 2*(lane>=16) + v
//   B (4x16): lane&15 = N col;  VGPR v holds K = 2*(lane>=16) + v
//   D (16x16): VGPR v, lane l:  M = v + 8*(lane>=16), N = lane&15
// ---------------------------------------------------------------------------
__global__ void delta_kernel(const float* __restrict__ prompt_sim,
                             const float* __restrict__ prompt_degree,
                             const float* __restrict__ prompt_other,
                             const float* __restrict__ readout,
                             const float* __restrict__ in_w,   // (384,128) row-major
                             const float* __restrict__ in_b,   // (384)
                             const float* __restrict__ out_w,  // (128,128) row-major
                             const float* __restrict__ out_b,  // (128)
                             float* __restrict__ delta) {      // (4,128)
    __shared__ float rec[16][C];      // 8 KB
    __shared__ float qkv[16][3 * C];  // 24 KB
    __shared__ float out0[C];

    int tid = threadIdx.x;  // single wave: 0..31

    // Build the 16 rec rows: row r = case*4 + token; case bits: s=(c>>1)&1, d=c&1
    for (int idx = tid; idx < 16 * C; idx += 32) {
        int r = idx >> 7, ch = idx & (C - 1);
        int cse = r >> 2, tok = r & 3;
        bool sm = (cse >> 1) & 1, dm = cse & 1;
        float v;
        if (tok == 0)      v = readout[ch];
        else if (tok == 1) v = sm ? prompt_sim[ch] : -1.0f;
        else if (tok == 2) v = dm ? prompt_degree[ch] : -1.0f;
        else               v = (!sm && !dm) ? prompt_other[ch] : -1.0f;
        rec[r][ch] = v;
    }
    __syncthreads();

    // qkv = rec(16x128) @ in_proj_w^T(128x384) + b, via WMMA f32 16x16x4.
    // B[k][n] = in_proj_w[n][k]; both A and B fragments are contiguous float2 loads.
    int half = tid >> 4, nn = tid & 15;
    for (int nt = 0; nt < 24; ++nt) {
        v8f acc = {};
        const float* wrow = in_w + (size_t)(nt * 16 + nn) * C;
        #pragma unroll 4
        for (int k0 = 0; k0 < C; k0 += 4) {
            v2f a, b;
            a.x = rec[nn][k0 + 2 * half];
            a.y = rec[nn][k0 + 2 * half + 1];
            b.x = wrow[k0 + 2 * half];
            b.y = wrow[k0 + 2 * half + 1];
            acc = __builtin_amdgcn_wmma_f32_16x16x4_f32(
                false, a, false, b, (short)0, acc, false, false);
        }
        float bias = in_b[nt * 16 + nn];
        #pragma unroll
        for (int v = 0; v < 8; ++v)
            qkv[v + 8 * half][nt * 16 + nn] = acc[v] + bias;
    }
    __syncthreads();

    // Tiny attention per case: only query token 0 matters (final = out[:,0,:]).
    const float scale = 0.08838834764831845f;  // 1/sqrt(128)
    for (int cse = 0; cse < 4; ++cse) {
        int base = cse * 4;
        bool sm = (cse >> 1) & 1, dm = cse & 1;
        bool pad[4] = {false, !sm, !dm, (sm || dm)};  // token all-(-1) => key-padded

        float sj[4];
        #pragma unroll
        for (int j = 0; j < 4; ++j) {
            float p = 0.f;
            #pragma unroll
            for (int u = 0; u < 4; ++u) {
                int ch = tid * 4 + u;
                p += qkv[base][ch] * qkv[base + j][C + ch];  // q0 . k_j
            }
            #pragma unroll
            for (int off = 16; off; off >>= 1) p += __shfl_xor(p, off, 32);
            sj[j] = p * scale;
        }
        float mx = -1e30f;
        #pragma unroll
        for (int j = 0; j < 4; ++j) if (!pad[j]) mx = fmaxf(mx, sj[j]);
        float w[4], wsum = 0.f;
        #pragma unroll
        for (int j = 0; j < 4; ++j) {
            w[j] = pad[j] ? 0.f : expf(sj[j] - mx);
            wsum += w[j];
        }
        float inv = 1.f / wsum;

        #pragma unroll
        for (int u = 0; u < 4; ++u) {
            int ch = tid * 4 + u;
            float o = 0.f;
            #pragma unroll
            for (int j = 0; j < 4; ++j) o += w[j] * qkv[base + j][2 * C + ch];
            out0[ch] = o * inv;
        }
        __syncthreads();

        // delta[cse] = out0 @ out_proj_w^T + out_proj_b
        #pragma unroll
        for (int u = 0; u < 4; ++u) {
            int ch = tid * 4 + u;
            float acc = out_b[ch];
            const float* wr = out_w + (size_t)ch * C;
            for (int k = 0; k < C; ++k) acc += out0[k] * wr[k];
            delta[cse * C + ch] = acc;
        }
        __syncthreads();
    }
}

// ---------------------------------------------------------------------------
// Kernel 4: out = x + delta[case(node)]. One wave per node, float4 per lane.
// HBM traffic here (read x + write out = 102 MB) is the true roofline.
// ---------------------------------------------------------------------------
__global__ void apply_kernel(const float* __restrict__ x, const float* __restrict__ csum,
                             const float* __restrict__ deg, const float* __restrict__ delta,
                             float* __restrict__ out, int n) {
    int gid = blockIdx.x * blockDim.x + threadIdx.x;
    int node = gid >> 5;
    if (node >= n) return;
    int chunk = gid & 31;
    float dg = deg[node];
    float cs = csum[node] / dg;                    // deg==0 -> NaN -> (NaN<=t) false, matches JAX
    int cse = ((cs <= 0.2f) ? 2 : 0) | ((dg <= 3.0f) ? 1 : 0);
    float4 a = ((const float4*)(x + (size_t)node * C))[chunk];
    float4 b = ((const float4*)(delta + (size_t)cse * C))[chunk];
    float4 r = make_float4(a.x + b.x, a.y + b.y, a.z + b.z, a.w + b.w);
    ((float4*)(out + (size_t)node * C))[chunk] = r;
}

extern "C" void kernel_launch(void* const* d_in, const int* in_sizes, int n_in,
                              void* d_out, int out_size, void* d_ws, size_t ws_size,
                              hipStream_t stream) {
    const float* x             = (const float*)d_in[0];
    const float* prompt_sim    = (const float*)d_in[1];
    const float* prompt_degree = (const float*)d_in[2];
    const float* prompt_other  = (const float*)d_in[3];
    const float* readout       = (const float*)d_in[4];
    const float* in_w          = (const float*)d_in[5];
    const float* in_b          = (const float*)d_in[6];
    const float* out_w         = (const float*)d_in[7];
    const float* out_b         = (const float*)d_in[8];
    const int*   ei            = (const int*)d_in[9];

    int N = in_sizes[0] / C;
    int E = in_sizes[9] / 2;

    float* invn  = (float*)d_ws;       // N
    float* csum  = invn + N;           // N (must be zeroed every call)
    float* degf  = csum + N;           // N (must be zeroed every call)
    float* delta = degf + N;           // 4*C

    hipMemsetAsync(csum, 0, 2 * (size_t)N * sizeof(float), stream);

    norm_kernel <<<(N + 7) / 8, 256, 0, stream>>>(x, invn, N);
    edge_kernel <<<(E + 7) / 8, 256, 0, stream>>>(x, ei, invn, csum, degf, E);
    delta_kernel<<<1, 32, 0, stream>>>(prompt_sim, prompt_degree, prompt_other, readout,
                                       in_w, in_b, out_w, out_b, delta);
    apply_kernel<<<(N * 32 + 255) / 256, 256, 0, stream>>>(x, csum, degf, delta,
                                                           (float*)d_out, N);
}